// GraphFunctionFilters_37091337568908
// MI455X (gfx1250) — compile-verified
//
#include <hip/hip_runtime.h>

typedef __attribute__((ext_vector_type(2))) float v2f;
typedef __attribute__((ext_vector_type(8))) float v8f;

#define V_TOT 20000
#define NPTS  16
#define SDIM  3
#define FDIM  16
#define KDIM  12   // S*NF

__global__ __launch_bounds__(256) void gff_wmma_kernel(
    const float* __restrict__ space,    // (V,16,3)
    const float* __restrict__ feat,     // (V,16,16)
    const float* __restrict__ scalers,  // (2,3,4) -> flat [p*12 + k]
    float* __restrict__ out)            // (V,16,12)
{
    const int lane = threadIdx.x & 31;
    const int wave = threadIdx.x >> 5;
    const int v    = blockIdx.x * 8 + wave;
    if (v >= V_TOT) return;  // wave-uniform; EXEC stays all-1s past here

    const float* sp = space + (size_t)v * (NPTS * SDIM);
    const float* ft = feat  + (size_t)v * (NPTS * FDIM);

    // ---- E[k]: lane k (k<12) accumulates sum_n exp(d^2 * scaler[sign]) ----
    const int   k   = (lane < KDIM) ? lane : 0;   // clamp, no divergence
    const int   s   = k >> 2;
    const float scp = scalers[k];                 // scalers[0][s][nf]
    const float scm = scalers[KDIM + k];          // scalers[1][s][nf]
    const float sp0 = sp[s];                      // space[v,0,s]

    float acc = 0.f;
#pragma unroll
    for (int n = 0; n < NPTS; ++n) {
        float d  = sp[n * SDIM + s] - sp0;
        float sc = (d >= 0.f) ? scp : scm;
        acc += __expf(d * d * sc);
    }
    v2f B;
    B.x = (lane < KDIM) ? acc : 0.f;   // row K=0 of B, cols 12..15 zero
    B.y = 0.f;                         // rows K=1..3 zero (unused anyway)

    // ---- Fsum[f]: lane f (f = lane&15) sums feat[v,:,f] -------------------
    const int f  = lane & 15;
    float fs = 0.f;
#pragma unroll
    for (int n = 0; n < NPTS; ++n)
        fs += ft[n * FDIM + f];        // per-n: lanes cover 64B contiguous

    // ---- C[j] = -Fsum[j + 8*half] (replicated across N) -------------------
    const int half = lane >> 4;        // 0: rows 0..7, 1: rows 8..15
    v8f C;
#pragma unroll
    for (int j = 0; j < 8; ++j) {
        float t = __shfl(fs, j + 8 * half, 32);
        C[j] = -t;
    }

    // ---- A: feat[v,0,m] in lanes 0..15 of VGPR0 (K=0); rest zero ----------
    float f0 = ft[f];                  // feat[v,0,lane&15]
    v2f A;
    A.x = (lane < 16) ? f0 : 0.f;
    A.y = 0.f;

    // ---- D = A x B + C : one 16x16x4 f32 WMMA per vertex ------------------
    v8f D = __builtin_amdgcn_wmma_f32_16x16x4_f32(
        /*neg_a=*/false, A, /*neg_b=*/false, B,
        /*c_mod=*/(short)0, C, /*reuse_a=*/false, /*reuse_b=*/false);

    // ---- store: VGPR j, lane l -> (m = j + 8*half, col = l&15), col<12 ----
    float* op  = out + (size_t)v * (FDIM * KDIM);
    const int col = lane & 15;
    if (col < KDIM) {
#pragma unroll
        for (int j = 0; j < 8; ++j)
            op[(j + 8 * half) * KDIM + col] = D[j];
    }
}

extern "C" void kernel_launch(void* const* d_in, const int* in_sizes, int n_in,
                              void* d_out, int out_size, void* d_ws, size_t ws_size,
                              hipStream_t stream) {
    const float* space   = (const float*)d_in[0];
    const float* feat    = (const float*)d_in[1];
    const float* scalers = (const float*)d_in[2];
    float*       out     = (float*)d_out;

    const int wavesPerBlock = 8;                         // 256 threads, wave32
    const int blocks = (V_TOT + wavesPerBlock - 1) / wavesPerBlock;  // 2500
    gff_wmma_kernel<<<blocks, 256, 0, stream>>>(space, feat, scalers, out);
}